// MultiDirectionalSSM_83631603188090
// MI455X (gfx1250) — compile-verified
//
#include <hip/hip_runtime.h>
#include <math.h>

#define LQ 1024
#define DQ 256
#define NDIRQ 6
#define NLQ 2
#define HQ 8
#define PQ 64
#define NSQ 32
#define KCQ 4
#define DIQ 512
#define DINQ 1096
#define DINPQ 1104   // DIN padded to multiple of 16 for WMMA tiling
#define CONVQ 576
#define FDIMQ 1536

typedef __attribute__((ext_vector_type(2))) float v2f;
typedef __attribute__((ext_vector_type(8))) float v8f;

// ---------------------------------------------------------------------------
// 1) Stable lexsort per direction: key = (c[a0]<<22)|(c[a1]<<16)|(c[a2]<<10)|i
// ---------------------------------------------------------------------------
__global__ void mdssm_sort(const int* __restrict__ coords,
                           int* __restrict__ perm, int* __restrict__ inv) {
  __shared__ unsigned keys[LQ];
  const int dir = blockIdx.x;
  const int a0t[6] = {0,0,1,1,2,2};
  const int a1t[6] = {1,2,0,2,0,1};
  const int a2t[6] = {2,1,2,0,1,0};
  const int a0 = a0t[dir], a1 = a1t[dir], a2 = a2t[dir];
  for (int i = threadIdx.x; i < LQ; i += blockDim.x) {
    unsigned c0 = (unsigned)coords[i*4 + a0];
    unsigned c1 = (unsigned)coords[i*4 + a1];
    unsigned c2 = (unsigned)coords[i*4 + a2];
    keys[i] = (((((c0 << 6) | c1) << 6) | c2) << 10) | (unsigned)i;
  }
  __syncthreads();
  for (int i = threadIdx.x; i < LQ; i += blockDim.x) {
    unsigned ki = keys[i];
    int r = 0;
    for (int j = 0; j < LQ; ++j) r += (keys[j] < ki) ? 1 : 0;
    perm[dir*LQ + r] = i;
    inv[dir*LQ + i]  = r;
  }
}

// ---------------------------------------------------------------------------
// 2) Gather x[dir][t] = vectors[perm[dir][t]]
// ---------------------------------------------------------------------------
__global__ void mdssm_gather(const float* __restrict__ v, const int* __restrict__ perm,
                             float* __restrict__ x) {
  int dir = blockIdx.y, t = blockIdx.x, d = threadIdx.x;
  x[((long long)(dir*LQ + t))*DQ + d] = v[(long long)perm[dir*LQ + t]*DQ + d];
}

// ---------------------------------------------------------------------------
// 3) LayerNorm over D=256
// ---------------------------------------------------------------------------
__global__ void mdssm_ln(const float* __restrict__ x, const float* __restrict__ g,
                         const float* __restrict__ b, float* __restrict__ out, int layer) {
  int dir = blockIdx.y, t = blockIdx.x, d = threadIdx.x;
  long long row = (long long)(dir*LQ + t) * DQ;
  float v = x[row + d];
  float s = v, s2 = v*v;
#pragma unroll
  for (int o = 16; o > 0; o >>= 1) { s += __shfl_xor(s, o, 32); s2 += __shfl_xor(s2, o, 32); }
  __shared__ float rs[8], rs2[8];
  if ((threadIdx.x & 31) == 0) { rs[threadIdx.x >> 5] = s; rs2[threadIdx.x >> 5] = s2; }
  __syncthreads();
  float S = 0.f, S2 = 0.f;
  for (int w = 0; w < 8; ++w) { S += rs[w]; S2 += rs2[w]; }
  float mu  = S * (1.0f / DQ);
  float var = S2 * (1.0f / DQ) - mu * mu;
  float r   = rsqrtf(var + 1e-5f);
  long long pb = (long long)(dir*NLQ + layer) * DQ;
  out[row + d] = (v - mu) * r * g[pb + d] + b[pb + d];
}

// ---------------------------------------------------------------------------
// 4) fp32 WMMA GEMM: Out[M x Ncols] = A[M x K] @ W[K x Ncols]   (per-batch z)
//    mode 0: store    mode 1: += Res (residual)   mode 2: +bias, exact GELU
//    Tail columns handled by clamped address + x0/x1 mask multiply -> no EXEC
//    manipulation inside the K loop; pointer-increment addressing.
// ---------------------------------------------------------------------------
__global__ void mdssm_gemm(const float* __restrict__ A, long long sAb,
                           const float* __restrict__ W, long long sWb,
                           float* Out, long long sOb,
                           const float* Res, long long sRb,
                           int Kd, int Ncols, int ldA, int ldW, int ldOut, int mode) {
  int z = blockIdx.z;
  A   += (long long)z * sAb;
  W   += (long long)z * sWb;
  Out += (long long)z * sOb;
  if (Res) Res += (long long)z * sRb;

  int waveId = threadIdx.x >> 5;
  int lane   = threadIdx.x & 31;
  int n0 = (blockIdx.x * 4 + (waveId & 3)) * 16;
  int m0 = (blockIdx.y * 2 + (waveId >> 2)) * 16;
  if (n0 >= Ncols) return;               // uniform per wave: EXEC stays full

  int am  = m0 + (lane & 15);            // A fragment: M = lane%16
  int kof = (lane >> 4) << 1;            // K sub-offset: 0 (lanes<16) / 2 (lanes>=16)
  int bn  = n0 + (lane & 15);            // B/D column
  bool bok = bn < Ncols;
  int bnc = bok ? bn : (Ncols - 1);      // clamped, always-valid column
  float bmask = bok ? 1.0f : 0.0f;       // zero out contributions of pad columns

  const float* Ap = A + (long long)am * ldA + kof;   // 8B aligned (ldA even, kof even)
  const float* Wp = W + (long long)kof * ldW + bnc;
  const long long wstep = 4ll * ldW;

  v8f c = {};
  for (int k0 = 0; k0 < Kd; k0 += 4) {
    v2f a = *(const v2f*)Ap;             // K = k0+kof, k0+kof+1
    v2f b;
    b.x = Wp[0]   * bmask;               // row K = k0+kof
    b.y = Wp[ldW] * bmask;               // row K = k0+kof+1
    c = __builtin_amdgcn_wmma_f32_16x16x4_f32(false, a, false, b, (short)0, c,
                                              false, false);
    Ap += 4;
    Wp += wstep;
  }
  if (bok) {
    int mb = m0 + ((lane >> 4) << 3);    // lanes>=16 hold rows M+8..M+15
#pragma unroll
    for (int r = 0; r < 8; ++r) {
      long long oi = (long long)(mb + r) * ldOut + bn;
      float v = c[r];
      if (mode == 1)      v += Res[oi];
      else if (mode == 2) { v += Res[bn]; v = 0.5f * v * (1.0f + erff(v * 0.70710678118654752f)); }
      Out[oi] = v;
    }
  }
}

// ---------------------------------------------------------------------------
// 5) Causal depthwise conv (K=4) + SiLU; softplus(dt); dA = exp(-exp(A_log)*dt)
// ---------------------------------------------------------------------------
__global__ void mdssm_conv(const float* __restrict__ zx,
                           const float* __restrict__ conv_w, const float* __restrict__ conv_b,
                           const float* __restrict__ dt_bias, const float* __restrict__ A_log,
                           float* __restrict__ xc, float* __restrict__ dtv,
                           float* __restrict__ dAv, int layer) {
  int dir = blockIdx.y, t = blockIdx.x, c = threadIdx.x;
  long long pb = (long long)(dir*NLQ + layer);
  const float* w = conv_w + (pb*CONVQ + c) * KCQ;
  float acc = conv_b[pb*CONVQ + c];
#pragma unroll
  for (int j = 0; j < KCQ; ++j) {
    int tt = t - (KCQ - 1) + j;          // block-uniform guard -> scalar branch
    if (tt >= 0) acc += zx[((long long)(dir*LQ + tt))*DINPQ + DIQ + c] * w[j];
  }
  float s = acc / (1.0f + expf(-acc));   // SiLU
  xc[((long long)(dir*LQ + t))*CONVQ + c] = s;
  if (c < HQ) {
    float raw = zx[((long long)(dir*LQ + t))*DINPQ + (DINQ - HQ) + c] + dt_bias[pb*HQ + c];
    float dt  = (raw > 20.0f) ? raw : log1pf(expf(raw));   // softplus
    dtv[(dir*LQ + t)*HQ + c] = dt;
    dAv[(dir*LQ + t)*HQ + c] = expf(-expf(A_log[pb*HQ + c]) * dt);
  }
}

// ---------------------------------------------------------------------------
// 6) Sequential SSM scan -> raw y only (gating moved to gate+rms kernel).
//    Block per (head, dir); thread: n = lane, p-octet = wave id.
//    x-values loaded as two b128 loads; y written as one coalesced masked
//    store (lanes 0..7) after a cndmask selection chain.
// ---------------------------------------------------------------------------
__global__ void mdssm_scan(const float* __restrict__ xc, const float* __restrict__ dtv,
                           const float* __restrict__ dAv, float* __restrict__ yout,
                           int layer) {
  int h = blockIdx.x, dir = blockIdx.y;
  int lane = threadIdx.x & 31;   // state index n
  int prow = threadIdx.x >> 5;   // p = prow*8 + k
  float hs[8];
#pragma unroll
  for (int k = 0; k < 8; ++k) hs[k] = 0.0f;

  for (int t = 0; t < LQ; ++t) {
    long long rowc = (long long)(dir*LQ + t) * CONVQ;
    float dA_t = dAv[(dir*LQ + t)*HQ + h];
    float dtB  = dtv[(dir*LQ + t)*HQ + h] * xc[rowc + DIQ + lane];
    float Cn   = xc[rowc + DIQ + NSQ + lane];
    float4 x0 = *(const float4*)(xc + rowc + h*PQ + prow*8);      // 16B aligned
    float4 x1 = *(const float4*)(xc + rowc + h*PQ + prow*8 + 4);
    float xv[8] = {x0.x, x0.y, x0.z, x0.w, x1.x, x1.y, x1.z, x1.w};
    float val = 0.0f;
#pragma unroll
    for (int k = 0; k < 8; ++k) {
      hs[k] = dA_t * hs[k] + dtB * xv[k];
      float yy = hs[k] * Cn;
#pragma unroll
      for (int o = 16; o > 0; o >>= 1) yy += __shfl_xor(yy, o, 32);
      val = (lane == k) ? yy : val;      // cndmask chain, no divergence
    }
    if (lane < 8)                        // one coalesced 32B masked store
      yout[(long long)(dir*LQ + t)*DIQ + h*PQ + prow*8 + lane] = val;
  }
}

// ---------------------------------------------------------------------------
// 7) Gate + RMSNorm (in place on u): u = (y + Dp*xs)*silu(z), then RMS*g.
// ---------------------------------------------------------------------------
__global__ void mdssm_gaterms(const float* __restrict__ xc, const float* __restrict__ zx,
                              const float* __restrict__ Dp, const float* __restrict__ g,
                              float* u, int layer) {
  int dir = blockIdx.y, t = blockIdx.x, d = threadIdx.x;
  long long row  = (long long)(dir*LQ + t) * DIQ;
  long long rowc = (long long)(dir*LQ + t) * CONVQ;
  long long rowz = (long long)(dir*LQ + t) * DINPQ;
  long long pb   = (long long)(dir*NLQ + layer);
  float vals[2];
#pragma unroll
  for (int j = 0; j < 2; ++j) {
    int cidx = d + j*DQ;                 // 0..511; h = cidx>>6, p = cidx&63
    float y  = u[row + cidx];
    float xv = xc[rowc + cidx];
    float zv = zx[rowz + cidx];
    float dp = Dp[pb*HQ + (cidx >> 6)];
    vals[j] = (y + dp * xv) * (zv / (1.0f + expf(-zv)));
  }
  float s = vals[0]*vals[0] + vals[1]*vals[1];
#pragma unroll
  for (int o = 16; o > 0; o >>= 1) s += __shfl_xor(s, o, 32);
  __shared__ float rs[8];
  if ((threadIdx.x & 31) == 0) rs[threadIdx.x >> 5] = s;
  __syncthreads();
  float S = 0.f;
  for (int w = 0; w < 8; ++w) S += rs[w];
  float scale = rsqrtf(S * (1.0f / DIQ) + 1e-5f);
  u[row + d]      = vals[0] * scale * g[pb*DIQ + d];
  u[row + DQ + d] = vals[1] * scale * g[pb*DIQ + DQ + d];
}

// ---------------------------------------------------------------------------
// 8) Unsort + concat: md[o][dir*D + d] = x[dir][inv[dir][o]][d]
// ---------------------------------------------------------------------------
__global__ void mdssm_unsort(const float* __restrict__ x, const int* __restrict__ inv,
                             float* __restrict__ md) {
  int dir = blockIdx.y, o = blockIdx.x, d = threadIdx.x;
  int t = inv[dir*LQ + o];
  md[(long long)o*FDIMQ + dir*DQ + d] = x[((long long)dir*LQ + t)*DQ + d];
}

// ---------------------------------------------------------------------------
// 9) Fuse LayerNorm over 1536
// ---------------------------------------------------------------------------
__global__ void mdssm_fuseln(const float* __restrict__ md, const float* __restrict__ g,
                             const float* __restrict__ b, float* __restrict__ out) {
  int t = blockIdx.x, d = threadIdx.x;
  long long row = (long long)t * FDIMQ;
  float vals[6]; float s = 0.f, s2 = 0.f;
#pragma unroll
  for (int j = 0; j < 6; ++j) {
    float v = md[row + j*DQ + d];
    vals[j] = v; s += v; s2 += v*v;
  }
#pragma unroll
  for (int o = 16; o > 0; o >>= 1) { s += __shfl_xor(s, o, 32); s2 += __shfl_xor(s2, o, 32); }
  __shared__ float rs[8], rs2[8];
  if ((threadIdx.x & 31) == 0) { rs[threadIdx.x >> 5] = s; rs2[threadIdx.x >> 5] = s2; }
  __syncthreads();
  float S = 0.f, S2 = 0.f;
  for (int w = 0; w < 8; ++w) { S += rs[w]; S2 += rs2[w]; }
  float mu  = S * (1.0f / FDIMQ);
  float var = S2 * (1.0f / FDIMQ) - mu * mu;
  float r   = rsqrtf(var + 1e-5f);
#pragma unroll
  for (int j = 0; j < 6; ++j)
    out[row + j*DQ + d] = (vals[j] - mu) * r * g[j*DQ + d] + b[j*DQ + d];
}

// ---------------------------------------------------------------------------
extern "C" void kernel_launch(void* const* d_in, const int* in_sizes, int n_in,
                              void* d_out, int out_size, void* d_ws, size_t ws_size,
                              hipStream_t stream) {
  const float* vectors  = (const float*)d_in[0];
  const int*   coords   = (const int*)  d_in[1];
  const float* ln_g     = (const float*)d_in[2];
  const float* ln_b     = (const float*)d_in[3];
  const float* Win      = (const float*)d_in[4];
  const float* conv_w   = (const float*)d_in[5];
  const float* conv_b   = (const float*)d_in[6];
  const float* dt_bias  = (const float*)d_in[7];
  const float* A_log    = (const float*)d_in[8];
  const float* Dp       = (const float*)d_in[9];
  const float* rms_g    = (const float*)d_in[10];
  const float* Wout     = (const float*)d_in[11];
  const float* fuse_g   = (const float*)d_in[12];
  const float* fuse_bb  = (const float*)d_in[13];
  const float* fuse_W   = (const float*)d_in[14];
  const float* fuse_b   = (const float*)d_in[15];
  float* out = (float*)d_out;

  char* ws = (char*)d_ws;
  size_t off = 0;
  auto carve = [&](size_t bytes) -> void* {
    void* p = ws + off;
    off = (off + bytes + 255) & ~(size_t)255;
    return p;
  };
  int*   perm = (int*)  carve((size_t)NDIRQ*LQ*4);
  int*   inv  = (int*)  carve((size_t)NDIRQ*LQ*4);
  float* x    = (float*)carve((size_t)NDIRQ*LQ*DQ*4);
  float* xn   = (float*)carve((size_t)NDIRQ*LQ*DQ*4);
  float* zx   = (float*)carve((size_t)NDIRQ*LQ*DINPQ*4);
  float* xc   = (float*)carve((size_t)NDIRQ*LQ*CONVQ*4);
  float* dtv  = (float*)carve((size_t)NDIRQ*LQ*HQ*4);
  float* dAv  = (float*)carve((size_t)NDIRQ*LQ*HQ*4);
  float* u    = (float*)carve((size_t)NDIRQ*LQ*DIQ*4);
  float* md   = (float*)carve((size_t)LQ*FDIMQ*4);
  float* mdn  = (float*)carve((size_t)LQ*FDIMQ*4);
  (void)ws_size; (void)in_sizes; (void)n_in; (void)out_size;

  mdssm_sort<<<NDIRQ, 256, 0, stream>>>(coords, perm, inv);
  mdssm_gather<<<dim3(LQ, NDIRQ), DQ, 0, stream>>>(vectors, perm, x);

  for (int l = 0; l < NLQ; ++l) {
    mdssm_ln<<<dim3(LQ, NDIRQ), DQ, 0, stream>>>(x, ln_g, ln_b, xn, l);
    // zxbcdt = xn @ Win   (M=1024, K=256, N=1096, out stride 1104)
    mdssm_gemm<<<dim3((DINPQ/16 + 3)/4, LQ/32, NDIRQ), 256, 0, stream>>>(
        xn, (long long)LQ*DQ,
        Win + (long long)l*DQ*DINQ, (long long)NLQ*DQ*DINQ,
        zx, (long long)LQ*DINPQ,
        nullptr, 0,
        DQ, DINQ, DQ, DINQ, DINPQ, 0);
    mdssm_conv<<<dim3(LQ, NDIRQ), CONVQ, 0, stream>>>(zx, conv_w, conv_b, dt_bias,
                                                      A_log, xc, dtv, dAv, l);
    mdssm_scan<<<dim3(HQ, NDIRQ), 256, 0, stream>>>(xc, dtv, dAv, u, l);
    mdssm_gaterms<<<dim3(LQ, NDIRQ), 256, 0, stream>>>(xc, zx, Dp, rms_g, u, l);
    // x = x + u @ Wout    (M=1024, K=512, N=256)
    mdssm_gemm<<<dim3((DQ/16)/4, LQ/32, NDIRQ), 256, 0, stream>>>(
        u, (long long)LQ*DIQ,
        Wout + (long long)l*DIQ*DQ, (long long)NLQ*DIQ*DQ,
        x, (long long)LQ*DQ,
        x, (long long)LQ*DQ,
        DIQ, DQ, DIQ, DQ, DQ, 1);
  }

  mdssm_unsort<<<dim3(LQ, NDIRQ), DQ, 0, stream>>>(x, inv, md);
  mdssm_fuseln<<<LQ, 256, 0, stream>>>(md, fuse_g, fuse_bb, mdn);
  // out = gelu(mdn @ fuse_W + fuse_b)   (M=1024, K=1536, N=256)
  mdssm_gemm<<<dim3((DQ/16)/4, LQ/32, 1), 256, 0, stream>>>(
      mdn, 0,
      fuse_W, 0,
      out, 0,
      fuse_b, 0,
      FDIMQ, DQ, FDIMQ, DQ, DQ, 2);
}